// SparseUpsample_3968549781707
// MI455X (gfx1250) — compile-verified
//
#include <hip/hip_runtime.h>

// ---- problem constants (from reference) ------------------------------------
#define NPTS    100000
#define IN_CH   256
#define OUT_CH  128
#define KVOL    8
#define M_TILES (NPTS / 16)          // 6250 exact
#define K_CHUNK 64                   // K rows staged in LDS per stage
#define LDS_STRIDE 136               // padded row stride (floats): 2*136 % 64 == 16
                                     // -> both ds reads of a B-frag hit 32 distinct banks

typedef float v2f __attribute__((ext_vector_type(2)));
typedef float v8f __attribute__((ext_vector_type(8)));

// ---------------------------------------------------------------------------
// GEMM: out[(row*8 + kblk)*128 + j] = sum_c feats[row,c] * W[kblk,c,j]
// One wave -> 16x128 output tile via 8 accumulators, fp32 WMMA 16x16x4.
// ---------------------------------------------------------------------------
__global__ __launch_bounds__(256) void sparse_upsample_gemm(
    const float* __restrict__ feats,   // [NPTS][IN_CH]
    const float* __restrict__ W,       // [KVOL][IN_CH][OUT_CH]
    float* __restrict__ out)           // [NPTS*KVOL][OUT_CH]
{
    __shared__ __attribute__((aligned(16))) float Bs[K_CHUNK * LDS_STRIDE];

    const int kblk = blockIdx.y;                 // which W_k (= 128-col strip)
    const int tid  = threadIdx.x;
    const int lane = tid & 31;
    const int wid  = tid >> 5;
    const int m16  = blockIdx.x * 8 + wid;       // 16-row tile index
    const bool valid = (m16 < M_TILES);

    const int rowA    = m16 * 16;
    const int halfSel = lane >> 4;               // 0: lanes 0-15, 1: lanes 16-31
    const int l15     = lane & 15;

    // Per-lane A base: 16x4 f32 A layout => lane<16 reads K,K+1 of row lane,
    // lane>=16 reads K+2,K+3 of row lane-16.
    const float* aBase =
        feats + (size_t)(valid ? (rowA + l15) : 0) * IN_CH + 2 * halfSel;

    v8f acc[8] = {};                             // 8 tiles x 8 VGPRs, zero-init

    // W_k slab for this block (contiguous [IN_CH][OUT_CH] floats)
    const float* wk = W + (size_t)kblk * IN_CH * OUT_CH;

    const int f4 = tid & 31;                     // float4 index within 128-wide row
    const int r0 = tid >> 5;                     // starting row for staging

    #pragma unroll 1
    for (int chunk = 0; chunk < IN_CH / K_CHUNK; ++chunk) {
        // ---- stage 64x128 floats of W_k into LDS (all waves participate) ----
        const float* wsrc = wk + (size_t)chunk * K_CHUNK * OUT_CH;
        #pragma unroll
        for (int r = r0; r < K_CHUNK; r += 8) {
            float4 v = ((const float4*)(wsrc + r * OUT_CH))[f4];
            *(float4*)&Bs[r * LDS_STRIDE + f4 * 4] = v;
        }
        if (chunk + 1 < IN_CH / K_CHUNK) {
            // hint next chunk toward the caches (global_prefetch_b8)
            __builtin_prefetch(wk + (size_t)(chunk + 1) * K_CHUNK * OUT_CH, 0, 0);
        }
        __syncthreads();

        // ---- compute: 16 K-steps of 4, 8 N-tiles each --------------------
        if (valid) {
            #pragma unroll 4
            for (int s = 0; s < K_CHUNK / 4; ++s) {
                const int kk = chunk * K_CHUNK + s * 4;
                v2f a = *(const v2f*)(aBase + kk);

                // B frag base: rows (4s + 2*halfSel) and +1, col l15 (+16 per tile)
                const int rb = (s * 4 + 2 * halfSel) * LDS_STRIDE + l15;
                #pragma unroll
                for (int t = 0; t < 8; ++t) {
                    v2f b;
                    b.x = Bs[rb + t * 16];
                    b.y = Bs[rb + t * 16 + LDS_STRIDE];
                    acc[t] = __builtin_amdgcn_wmma_f32_16x16x4_f32(
                        /*neg_a=*/false, a, /*neg_b=*/false, b,
                        /*c_mod=*/(short)0, acc[t],
                        /*reuse_a=*/false, /*reuse_b=*/false);
                }
            }
        }
        __syncthreads();
    }

    // ---- store: C layout VGPR p = rows {p, p+8}, N = lane ------------------
    if (valid) {
        #pragma unroll
        for (int t = 0; t < 8; ++t) {
            #pragma unroll
            for (int p = 0; p < 8; ++p) {
                const int row = rowA + p + 8 * halfSel;
                const size_t o =
                    ((size_t)row * KVOL + kblk) * OUT_CH + t * 16 + l15;
                out[o] = acc[t][p];
            }
        }
    }
}

// ---------------------------------------------------------------------------
// out_coords[(n*8+k), d] = 2*coords[n,d] + offs[k,d], offs in ij meshgrid order
// ---------------------------------------------------------------------------
__global__ __launch_bounds__(256) void sparse_upsample_coords(
    const int* __restrict__ coords,    // [NPTS][3]
    int* __restrict__ outc,            // [NPTS*KVOL][3]
    int total)
{
    const int i = blockIdx.x * blockDim.x + threadIdx.x;  // over NPTS*KVOL
    if (i >= total) return;
    const int n = i >> 3;
    const int k = i & 7;
    outc[i * 3 + 0] = coords[n * 3 + 0] * 2 + ((k >> 2) & 1);
    outc[i * 3 + 1] = coords[n * 3 + 1] * 2 + ((k >> 1) & 1);
    outc[i * 3 + 2] = coords[n * 3 + 2] * 2 + (k & 1);
}

// ---------------------------------------------------------------------------
extern "C" void kernel_launch(void* const* d_in, const int* in_sizes, int n_in,
                              void* d_out, int out_size, void* d_ws, size_t ws_size,
                              hipStream_t stream) {
    (void)in_sizes; (void)n_in; (void)out_size; (void)d_ws; (void)ws_size;

    const float* feats  = (const float*)d_in[0];   // [NPTS*IN_CH] f32
    const int*   coords = (const int*)  d_in[1];   // [NPTS*3]     i32
    const float* W      = (const float*)d_in[2];   // [KVOL*IN_CH*OUT_CH] f32

    float* out_feats  = (float*)d_out;                                   // [NPTS*KVOL*OUT_CH]
    int*   out_coords = (int*)(out_feats + (size_t)NPTS * KVOL * OUT_CH); // [NPTS*KVOL*3]

    dim3 grid((M_TILES + 7) / 8, KVOL);            // (782, 8)
    sparse_upsample_gemm<<<grid, 256, 0, stream>>>(feats, W, out_feats);

    const int total = NPTS * KVOL;                 // 800000
    sparse_upsample_coords<<<(total + 255) / 256, 256, 0, stream>>>(
        coords, out_coords, total);
}